// Mamba2StepWrapper_73100343378593
// MI455X (gfx1250) — compile-verified
//
#include <hip/hip_runtime.h>
#include <hip/hip_bf16.h>

// ---------------- model dimensions (from reference) ----------------
#define Lx 12
#define Bx 8
#define Dx 2048
#define Vx 32000
#define Hx 64
#define HDx 64
#define Gx 8
#define Nx 128
#define Kx 4
#define Ix (Hx * HDx)            // 4096
#define CONVx (Ix + 2 * Gx * Nx) // 6144
#define Px (Ix + CONVx + Hx)     // 10304
#define TMAXf 10000.0f

typedef __attribute__((ext_vector_type(2))) float v2f;
typedef __attribute__((ext_vector_type(8))) float v8f;

// ---------------- helpers ----------------
__device__ __forceinline__ float block_reduce_sum(float v, float* red) {
  for (int m = 16; m >= 1; m >>= 1) v += __shfl_xor(v, m, 32);
  int wid = threadIdx.x >> 5, lane = threadIdx.x & 31;
  if (lane == 0) red[wid] = v;
  __syncthreads();
  float t = (threadIdx.x < (blockDim.x >> 5)) ? red[threadIdx.x] : 0.f;
  if (wid == 0) {
    for (int m = 16; m >= 1; m >>= 1) t += __shfl_xor(t, m, 32);
    if (lane == 0) red[0] = t;
  }
  __syncthreads();
  return red[0];
}

__device__ __forceinline__ float silu_f(float x) { return x / (1.f + __expf(-x)); }

// ---------------- embedding gather ----------------
__global__ void embed_kernel(const int* __restrict__ ids, const float* __restrict__ emb,
                             float* __restrict__ h) {
  int b = blockIdx.x;
  int id = ids[b];
  const float* src = emb + (size_t)id * Dx;
  float* dst = h + (size_t)b * Dx;
  for (int i = threadIdx.x; i < Dx; i += blockDim.x) dst[i] = src[i];
}

// ---------------- RMSNorm (one block per batch row) ----------------
__global__ void rmsnorm_kernel(const float* __restrict__ x, const float* __restrict__ w,
                               float* __restrict__ out, int ncols) {
  __shared__ float red[32];
  int b = blockIdx.x;
  const float* xr = x + (size_t)b * ncols;
  float* orow = out + (size_t)b * ncols;
  float ss = 0.f;
  for (int i = threadIdx.x; i < ncols; i += blockDim.x) { float v = xr[i]; ss += v * v; }
  float tot = block_reduce_sum(ss, red);
  float inv = rsqrtf(tot / (float)ncols + 1e-5f);
  for (int i = threadIdx.x; i < ncols; i += blockDim.x) orow[i] = xr[i] * inv * w[i];
}

// ---------------- skinny GEMM with split-K: (8 x KD) @ (KD x N) -----------------
// One wave owns TWO 16-column tiles (32 cols) and one K-chunk of KD/SPLITK rows;
// partials go to a per-chunk buffer, summed by reduce_partials (deterministic).
// M = batch padded 8 -> 16; C row m depends only on A row m and rows 8..15 are
// discarded, so A rows 8..15 load as duplicates of rows 0..7 (branch-free).
// V_WMMA_F32_16X16X4_F32 fragment layouts (ISA 7.12.2):
//   A 16x4 : lanes 0-15 -> M=lane, K={0,1}; lanes 16-31 -> M=lane-16, K={2,3}
//   B 4x16 : lanes 0-15 -> N=lane, K={0,1}; lanes 16-31 -> N=lane-16, K={2,3}
//   C 16x16: VGPR g, lanes 0-15 -> M=g; lanes 16-31 -> M=g+8
template <int KD, int LDW, int SPLITK>
__global__ void gemm8_nn_wmma(const float* __restrict__ A, const float* __restrict__ W,
                              float* __restrict__ Part, int Ncols) {
  constexpr int KC = KD / SPLITK; // K-rows per chunk (multiple of 64)
  int wave = blockIdx.x * (blockDim.x >> 5) + (threadIdx.x >> 5);
  int tile = wave / SPLITK;
  int chunk = wave - tile * SPLITK;
  int col0 = tile * 32;
  if (col0 >= Ncols) return; // wave-uniform: surviving waves keep EXEC all-ones
  int lane = threadIdx.x & 31;
  int lid = lane & 15;
  int kb = (lane >> 4) * 2; // 0 for lanes 0-15, 2 for lanes 16-31
  const int k0 = chunk * KC;
  const float* arow = A + (size_t)(lid & 7) * KD + k0 + kb; // rows 8..15 dup 0..7
  const float* wp = W + (size_t)(k0 + kb) * LDW + col0 + lid;
  v8f c0 = {0.f, 0.f, 0.f, 0.f, 0.f, 0.f, 0.f, 0.f};
  v8f c1 = {0.f, 0.f, 0.f, 0.f, 0.f, 0.f, 0.f, 0.f};
  for (int kk = 0; kk < KC; kk += 64) {
    __builtin_prefetch(wp + (size_t)(kk + 128) * LDW, 0, 1); // speculative stream-ahead
#pragma unroll
    for (int ki = 0; ki < 64; ki += 4) {
      const int k = kk + ki;
      v2f a = *(const v2f*)(arow + k);        // hot in cache (tiny A)
      const float* w0 = wp + (size_t)k * LDW; // compile-time strides
      v2f b0, b1;
      b0.x = __builtin_nontemporal_load(w0);  // weights: single-use stream
      b0.y = __builtin_nontemporal_load(w0 + LDW);
      b1.x = __builtin_nontemporal_load(w0 + 16);
      b1.y = __builtin_nontemporal_load(w0 + LDW + 16);
      c0 = __builtin_amdgcn_wmma_f32_16x16x4_f32(false, a, false, b0, (short)0, c0,
                                                 false, false);
      c1 = __builtin_amdgcn_wmma_f32_16x16x4_f32(false, a, false, b1, (short)0, c1,
                                                 false, false);
    }
  }
  if (lane < 16) { // rows M=0..7 live in VGPRs 0..7 of lanes 0-15
#pragma unroll
    for (int g = 0; g < 8; ++g) {
      size_t o = ((size_t)chunk * Bx + g) * Ncols + col0 + lane;
      Part[o] = c0[g];
      Part[o + 16] = c1[g];
    }
  }
}

// Transposed-B variant for the LM head: logits[b,v] = sum_d A[b,d]*Wt[v,d]
template <int KD, int SPLITK>
__global__ void gemm8_nt_wmma(const float* __restrict__ A, const float* __restrict__ Wt,
                              float* __restrict__ Part, int Ncols) {
  constexpr int KC = KD / SPLITK;
  int wave = blockIdx.x * (blockDim.x >> 5) + (threadIdx.x >> 5);
  int tile = wave / SPLITK;
  int chunk = wave - tile * SPLITK;
  int col0 = tile * 32;
  if (col0 >= Ncols) return;
  int lane = threadIdx.x & 31;
  int lid = lane & 15;
  int kb = (lane >> 4) * 2;
  const int k0 = chunk * KC;
  const float* arow = A + (size_t)(lid & 7) * KD + k0 + kb;
  const float* wr0 = Wt + (size_t)(col0 + lid) * KD + k0 + kb; // 2 vocab rows/lane
  const float* wr1 = wr0 + (size_t)16 * KD;
  v8f c0 = {0.f, 0.f, 0.f, 0.f, 0.f, 0.f, 0.f, 0.f};
  v8f c1 = {0.f, 0.f, 0.f, 0.f, 0.f, 0.f, 0.f, 0.f};
  for (int kk = 0; kk < KC; kk += 64) {
    __builtin_prefetch(wr0 + kk + 512, 0, 1);
    __builtin_prefetch(wr1 + kk + 512, 0, 1);
#pragma unroll
    for (int ki = 0; ki < 64; ki += 4) {
      const int k = kk + ki;
      v2f a = *(const v2f*)(arow + k);
      v2f b0 = __builtin_nontemporal_load((const v2f*)(wr0 + k)); // b64 stream
      v2f b1 = __builtin_nontemporal_load((const v2f*)(wr1 + k));
      c0 = __builtin_amdgcn_wmma_f32_16x16x4_f32(false, a, false, b0, (short)0, c0,
                                                 false, false);
      c1 = __builtin_amdgcn_wmma_f32_16x16x4_f32(false, a, false, b1, (short)0, c1,
                                                 false, false);
    }
  }
  if (lane < 16) {
#pragma unroll
    for (int g = 0; g < 8; ++g) {
      size_t o = ((size_t)chunk * Bx + g) * Ncols + col0 + lane;
      Part[o] = c0[g];
      Part[o + 16] = c1[g];
    }
  }
}

// ---------------- deterministic split-K reduction ----------------
// Out[row, col] = (accum ? Out : 0) + sum_s Part[s][row][col], fixed order.
template <int SPLITK>
__global__ void reduce_partials(const float* __restrict__ Part, float* __restrict__ Out,
                                int ldo, int Ncols, int accum) {
  int t = blockIdx.x * blockDim.x + threadIdx.x;
  if (t >= Bx * Ncols) return;
  int row = t / Ncols, col = t - row * Ncols;
  float s = 0.f;
#pragma unroll
  for (int c = 0; c < SPLITK; ++c) s += Part[((size_t)c * Bx + row) * Ncols + col];
  size_t o = (size_t)row * ldo + col;
  Out[o] = accum ? (Out[o] + s) : s;
}

// ---------------- conv state shift + depthwise conv + SiLU ----------------
__global__ void conv_kernel(const float* __restrict__ cs, const float* __restrict__ proj,
                            const float* __restrict__ cw, const float* __restrict__ cb,
                            float* __restrict__ ncs, float* __restrict__ convout) {
  int t = blockIdx.x * blockDim.x + threadIdx.x;
  if (t >= Bx * CONVx) return;
  int b = t / CONVx, cch = t - b * CONVx;
  float4 s4 = *(const float4*)(cs + (size_t)t * Kx);
  float nin = proj[(size_t)b * Px + Ix + cch];
  float4 ns = make_float4(s4.y, s4.z, s4.w, nin);
  *(float4*)(ncs + (size_t)t * Kx) = ns;
  float4 w4 = *(const float4*)(cw + (size_t)cch * Kx);
  float s = ns.x * w4.x + ns.y * w4.y + ns.z * w4.z + ns.w * w4.w + cb[cch];
  convout[t] = silu_f(s);
}

// ---------------- dt = clip(softplus(dt_raw + bias)), dA = exp(dt * -exp(A_log)) ---
__global__ void dt_kernel(const float* __restrict__ proj, const float* __restrict__ db,
                          const float* __restrict__ al, float* __restrict__ dtbuf,
                          float* __restrict__ dAbuf) {
  int t = blockIdx.x * blockDim.x + threadIdx.x;
  if (t >= Bx * Hx) return;
  int b = t / Hx, hh = t - b * Hx;
  float x = proj[(size_t)b * Px + Ix + CONVx + hh] + db[hh];
  float sp = (x > 20.f) ? x : log1pf(__expf(x));
  sp = fminf(fmaxf(sp, 0.f), TMAXf);
  dtbuf[t] = sp;
  dAbuf[t] = __expf(sp * (-__expf(al[hh])));
}

// ---------------- SSM state update + readout (one wave per (b,h,hd)) ----------------
// Each lane owns 4 contiguous n (float4 / b128), wave covers a 512B contiguous run.
__global__ void ssm_kernel(const float* __restrict__ ss_in, const float* __restrict__ convout,
                           const float* __restrict__ dtbuf, const float* __restrict__ dAbuf,
                           const float* __restrict__ dp, float* __restrict__ ss_out,
                           float* __restrict__ y) {
  int wave = blockIdx.x * (blockDim.x >> 5) + (threadIdx.x >> 5);
  int lane = threadIdx.x & 31;
  int b = wave / (Hx * HDx);
  int rem = wave - b * Hx * HDx;
  int hh = rem / HDx;
  int hd = rem - hh * HDx;
  int g = hh / (Hx / Gx);
  float xv = convout[(size_t)b * CONVx + hh * HDx + hd];
  float dt = dtbuf[b * Hx + hh];
  float dA = dAbuf[b * Hx + hh];
  const float* Bp = convout + (size_t)b * CONVx + Ix + g * Nx + lane * 4;
  const float* Cp = Bp + Gx * Nx;
  size_t off = (size_t)wave * Nx + lane * 4; // (b,h,hd,n) flattened
  float4 s = *(const float4*)(ss_in + off);
  float4 Bn = *(const float4*)Bp;
  float4 Cn = *(const float4*)Cp;
  float dtx = dt * xv;
  float4 ns;
  ns.x = s.x * dA + dtx * Bn.x;
  ns.y = s.y * dA + dtx * Bn.y;
  ns.z = s.z * dA + dtx * Bn.z;
  ns.w = s.w * dA + dtx * Bn.w;
  *(float4*)(ss_out + off) = ns;
  float acc = ns.x * Cn.x + ns.y * Cn.y + ns.z * Cn.z + ns.w * Cn.w;
  for (int m = 16; m >= 1; m >>= 1) acc += __shfl_xor(acc, m, 32);
  if (lane == 0) y[(size_t)b * Ix + hh * HDx + hd] = acc + xv * dp[hh];
}

// ---------------- yg = y * silu(gate); out = rmsnorm(yg) * gn_w ----------------
__global__ void gated_rmsnorm_kernel(const float* __restrict__ y, const float* __restrict__ proj,
                                     const float* __restrict__ gw, float* __restrict__ out) {
  __shared__ float red[32];
  int b = blockIdx.x;
  const float* yr = y + (size_t)b * Ix;
  const float* gr = proj + (size_t)b * Px; // gate occupies proj[:, 0:I]
  float* orow = out + (size_t)b * Ix;
  float ss = 0.f;
  for (int i = threadIdx.x; i < Ix; i += blockDim.x) {
    float v = yr[i] * silu_f(gr[i]);
    orow[i] = v;
    ss += v * v;
  }
  float tot = block_reduce_sum(ss, red);
  float inv = rsqrtf(tot / (float)Ix + 1e-5f);
  for (int i = threadIdx.x; i < Ix; i += blockDim.x) orow[i] = orow[i] * inv * gw[i];
}

// ---------------- launch ----------------
extern "C" void kernel_launch(void* const* d_in, const int* in_sizes, int n_in,
                              void* d_out, int out_size, void* d_ws, size_t ws_size,
                              hipStream_t stream) {
  const int* ids = (const int*)d_in[0];
  const float* conv_states = (const float*)d_in[1];
  const float* ssm_states = (const float*)d_in[2];
  const float* emb = (const float*)d_in[3];
  const float* norm_w = (const float*)d_in[4];
  const float* in_proj_w = (const float*)d_in[5];
  const float* conv_w = (const float*)d_in[6];
  const float* conv_b = (const float*)d_in[7];
  const float* dt_bias = (const float*)d_in[8];
  const float* A_log = (const float*)d_in[9];
  const float* D_param = (const float*)d_in[10];
  const float* gn_w = (const float*)d_in[11];
  const float* out_proj_w = (const float*)d_in[12];
  const float* norm_f_w = (const float*)d_in[13];
  const float* lm_head_w = (const float*)d_in[14];

  float* outp = (float*)d_out;
  float* logits = outp;                                     // B*V
  float* new_conv = logits + (size_t)Bx * Vx;               // L*B*CONV*K
  float* new_ssm = new_conv + (size_t)Lx * Bx * CONVx * Kx; // L*B*H*HD*N

  float* ws = (float*)d_ws;
  float* h = ws;       ws += (size_t)Bx * Dx;
  float* hn = ws;      ws += (size_t)Bx * Dx;
  float* proj = ws;    ws += (size_t)Bx * Px;
  float* convout = ws; ws += (size_t)Bx * CONVx;
  float* dtbuf = ws;   ws += (size_t)Bx * Hx;
  float* dAbuf = ws;   ws += (size_t)Bx * Hx;
  float* ybuf = ws;    ws += (size_t)Bx * Ix;
  float* y2 = ws;      ws += (size_t)Bx * Ix;
  float* hf = ws;      ws += (size_t)Bx * Dx;
  float* part = ws;    ws += (size_t)32 * Bx * Vx / 8; // 4 * Bx * Vx >= all partials

  embed_kernel<<<Bx, 256, 0, stream>>>(ids, emb, h);

  constexpr int S_IN = 8;   // in_proj: 322 tiles * 8  = 2576 waves
  constexpr int S_OUT = 32; // out_proj: 64 tiles * 32 = 2048 waves
  constexpr int S_LM = 4;   // lm_head: 1000 tiles * 4 = 4000 waves

  for (int l = 0; l < Lx; ++l) {
    rmsnorm_kernel<<<Bx, 512, 0, stream>>>(h, norm_w + (size_t)l * Dx, hn, Dx);

    // in_proj: (8 x 2048) @ (2048 x 10304) -> proj (split-K partials + reduce)
    {
      int waves = (Px / 32) * S_IN;
      gemm8_nn_wmma<Dx, Px, S_IN><<<(waves + 7) / 8, 256, 0, stream>>>(
          hn, in_proj_w + (size_t)l * Dx * Px, part, Px);
      reduce_partials<S_IN><<<(Bx * Px + 255) / 256, 256, 0, stream>>>(
          part, proj, Px, Px, 0);
    }

    conv_kernel<<<(Bx * CONVx + 255) / 256, 256, 0, stream>>>(
        conv_states + (size_t)l * Bx * CONVx * Kx, proj,
        conv_w + (size_t)l * CONVx * Kx, conv_b + (size_t)l * CONVx,
        new_conv + (size_t)l * Bx * CONVx * Kx, convout);

    dt_kernel<<<(Bx * Hx + 255) / 256, 256, 0, stream>>>(
        proj, dt_bias + (size_t)l * Hx, A_log + (size_t)l * Hx, dtbuf, dAbuf);

    ssm_kernel<<<(Bx * Hx * HDx) / 8, 256, 0, stream>>>(
        ssm_states + (size_t)l * Bx * Hx * HDx * Nx, convout, dtbuf, dAbuf,
        D_param + (size_t)l * Hx, new_ssm + (size_t)l * Bx * Hx * HDx * Nx, ybuf);

    gated_rmsnorm_kernel<<<Bx, 1024, 0, stream>>>(ybuf, proj, gn_w + (size_t)l * Ix, y2);

    // out_proj + residual into h: (8 x 4096) @ (4096 x 2048)
    {
      int waves = (Dx / 32) * S_OUT;
      gemm8_nn_wmma<Ix, Dx, S_OUT><<<(waves + 7) / 8, 256, 0, stream>>>(
          y2, out_proj_w + (size_t)l * Ix * Dx, part, Dx);
      reduce_partials<S_OUT><<<(Bx * Dx + 255) / 256, 256, 0, stream>>>(
          part, h, Dx, Dx, 1);
    }
  }

  rmsnorm_kernel<<<Bx, 512, 0, stream>>>(h, norm_f_w, hf, Dx);

  // lm_head: logits[b,v] = sum_d hf[b,d] * lm_head_w[v,d]
  {
    int waves = (Vx / 32) * S_LM;
    gemm8_nt_wmma<Dx, S_LM><<<(waves + 7) / 8, 256, 0, stream>>>(
        hf, lm_head_w, part, Vx);
    reduce_partials<S_LM><<<(Bx * Vx + 255) / 256, 256, 0, stream>>>(
        part, logits, Vx, Vx, 0);
  }
}